// HausdorffERLoss_40312563040871
// MI455X (gfx1250) — compile-verified
//
#include <hip/hip_runtime.h>

// HausdorffERLoss on MI455X (gfx1250).
// Memory-bound: 10 x (read 64MB + write 64MB) ping-pong; both buffers fit the
// 192MB L2. Per-sample min/max/sum reductions replace materializing the
// normalized field; normalization folded into the next iteration's loads.
// 3x3 cross conv done per 16x16 tile as T*Bh + Bv*T via 8 chained
// v_wmma_f32_16x16x4_f32, band matrices are lane-computed constants.

#define HW_W 1024
#define HW_H 1024
#define NB   16
#define NPIX (HW_H * HW_W)

typedef float v2f __attribute__((ext_vector_type(2)));
typedef float v8f __attribute__((ext_vector_type(8)));

template<bool FIRST>
__device__ __forceinline__ float ldval(const float* __restrict__ p0,
                                       const float* __restrict__ p1,
                                       float mn, float inv, int idx) {
  if (FIRST) { float d = p0[idx] - p1[idx]; return d * d; }  // bound = (pred-target)^2
  return (p0[idx] - mn) * inv;                               // normalize on load
}

template<bool FIRST, bool LAST>
__global__ __launch_bounds__(256) void erode_step(
    const float* __restrict__ p0, const float* __restrict__ p1,
    const float* __restrict__ params,   // per-sample {mn, 1/ptp}
    float* __restrict__ dst,
    float* __restrict__ rmin, float* __restrict__ rmax, double* __restrict__ sumd)
{
  const int lane = threadIdx.x & 31;
  const int wid  = threadIdx.x >> 5;
  const int bx   = blockIdx.x;
  const int sample   = bx >> 9;            // 512 blocks per image
  const int rest     = bx & 511;
  const int tile_row = rest >> 3;          // 0..63
  const int tile_col = ((rest & 7) << 3) + wid;  // 8 adjacent tiles per block
  const int r0 = tile_row << 4;
  const int c0 = tile_col << 4;
  const int base = sample * NPIX;

  float mn = 0.f, inv = 1.f;
  if (!FIRST) { mn = params[2 * sample]; inv = params[2 * sample + 1]; }

  const int h  = lane >> 4;   // half-wave index
  const int mi = lane & 15;   // row (A-frag) / col (B-frag) index

  v8f c = {0.f, 0.f, 0.f, 0.f, 0.f, 0.f, 0.f, 0.f};

  // H3 = T x Bh : A-frag = tile (two adjacent columns per lane -> b64-able),
  //               B-frag = tridiag(1,1,1) lane-computed.
  #pragma unroll
  for (int q = 0; q < 4; ++q) {
    const int k0 = 4 * q + 2 * h;
    v2f a, bh;
    a.x = ldval<FIRST>(p0, p1, mn, inv, base + (r0 + mi) * HW_W + (c0 + k0));
    a.y = ldval<FIRST>(p0, p1, mn, inv, base + (r0 + mi) * HW_W + (c0 + k0 + 1));
    const int d0 = k0 - mi, d1 = k0 + 1 - mi;
    bh.x = (d0 >= -1 && d0 <= 1) ? 1.f : 0.f;
    bh.y = (d1 >= -1 && d1 <= 1) ? 1.f : 0.f;
    c = __builtin_amdgcn_wmma_f32_16x16x4_f32(false, a, false, bh, (short)0, c, false, false);
  }
  // V3 = Bv x T : A-frag = off-diagonal band, B-frag = tile.
  #pragma unroll
  for (int q = 0; q < 4; ++q) {
    const int k0 = 4 * q + 2 * h;
    v2f bv, b;
    const int d0 = mi - k0, d1 = mi - (k0 + 1);
    bv.x = (d0 == 1 || d0 == -1) ? 1.f : 0.f;
    bv.y = (d1 == 1 || d1 == -1) ? 1.f : 0.f;
    b.x = ldval<FIRST>(p0, p1, mn, inv, base + (r0 + k0) * HW_W + (c0 + mi));
    b.y = ldval<FIRST>(p0, p1, mn, inv, base + (r0 + k0 + 1) * HW_W + (c0 + mi));
    c = __builtin_amdgcn_wmma_f32_16x16x4_f32(false, bv, false, b, (short)0, c, false, false);
  }

  // Halo fix-ups (image-border zero padding handled by omission).
  // C-layout: lane holds (M = v + 8h, N = mi).
  if (mi == 0 && c0 > 0) {
    #pragma unroll
    for (int v = 0; v < 8; ++v)
      c[v] += ldval<FIRST>(p0, p1, mn, inv, base + (r0 + v + 8 * h) * HW_W + (c0 - 1));
  }
  if (mi == 15 && c0 + 16 < HW_W) {
    #pragma unroll
    for (int v = 0; v < 8; ++v)
      c[v] += ldval<FIRST>(p0, p1, mn, inv, base + (r0 + v + 8 * h) * HW_W + (c0 + 16));
  }
  if (h == 0 && r0 > 0)
    c[0] += ldval<FIRST>(p0, p1, mn, inv, base + (r0 - 1) * HW_W + (c0 + mi));
  if (h == 1 && r0 + 16 < HW_H)
    c[7] += ldval<FIRST>(p0, p1, mn, inv, base + (r0 + 16) * HW_W + (c0 + mi));

  // ero = max(0.2*cross - 0.5, 0); write raw field + local reductions.
  float lmn = __int_as_float(0x7F800000);
  float lmx = 0.f, lsm = 0.f;
  #pragma unroll
  for (int v = 0; v < 8; ++v) {
    float e = fmaxf(fmaf(0.2f, c[v], -0.5f), 0.f);
    if (!LAST) dst[base + (r0 + v + 8 * h) * HW_W + (c0 + mi)] = e;
    lmn = fminf(lmn, e); lmx = fmaxf(lmx, e); lsm += e;
  }

  // wave32 reduction
  #pragma unroll
  for (int off = 16; off >= 1; off >>= 1) {
    lmn = fminf(lmn, __shfl_xor(lmn, off, 32));
    lmx = fmaxf(lmx, __shfl_xor(lmx, off, 32));
    lsm += __shfl_xor(lsm, off, 32);
  }

  __shared__ float smn[8], smx[8], ssm[8];
  if (lane == 0) { smn[wid] = lmn; smx[wid] = lmx; ssm[wid] = lsm; }
  __syncthreads();
  if (threadIdx.x == 0) {
    float bmn = smn[0], bmx = smx[0], bsm = ssm[0];
    #pragma unroll
    for (int i = 1; i < 8; ++i) {
      bmn = fminf(bmn, smn[i]); bmx = fmaxf(bmx, smx[i]); bsm += ssm[i];
    }
    // ero >= 0, so IEEE bit pattern order == value order for int atomics.
    atomicMin((int*)&rmin[sample], __float_as_int(bmn));
    atomicMax((int*)&rmax[sample], __float_as_int(bmx));
    atomicAdd(&sumd[sample], (double)bsm);
  }
}

__global__ void init_state(float* rmin, float* rmax, double* sumd,
                           float* params, double* accum) {
  const int t = threadIdx.x;
  if (t < NB) {
    rmin[t] = __int_as_float(0x7F800000);
    rmax[t] = 0.f;
    sumd[t] = 0.0;
    params[2 * t] = 0.f;
    params[2 * t + 1] = 1.f;
  }
  if (t == 0) accum[0] = 0.0;
}

__global__ void finalize_iter(float* rmin, float* rmax, double* sumd,
                              float* params, double* accum, float* out,
                              int k, int last) {
  const int s = threadIdx.x;
  double contrib = 0.0;
  if (s < NB) {
    const float mnf = rmin[s], mxf = rmax[s];
    const double sum = sumd[s];
    const float ptp = mxf - mnf;
    float pmn, pinv;
    double ns;
    if (ptp != 0.f) {
      pinv = 1.f / ptp;
      ns = (sum - (double)NPIX * (double)mnf) / (double)ptp;  // sum of normalized ero
      pmn = mnf;
    } else {
      ns = sum; pmn = 0.f; pinv = 1.f;   // matches jnp.where(ptp==0) branch
    }
    params[2 * s] = pmn;
    params[2 * s + 1] = pinv;
    double w = (double)(k + 1); w *= w;  // (k+1)^ALPHA, ALPHA=2
    contrib = ns * w;
    // reset reduction slots for the next iteration
    rmin[s] = __int_as_float(0x7F800000);
    rmax[s] = 0.f;
    sumd[s] = 0.0;
  }
  #pragma unroll
  for (int off = 16; off >= 1; off >>= 1)
    contrib += __shfl_xor(contrib, off, 32);
  if (s == 0) {
    const double a = accum[0] + contrib;
    accum[0] = a;
    if (last) out[0] = (float)(a * (1.0 / ((double)NB * (double)NPIX)));
  }
}

extern "C" void kernel_launch(void* const* d_in, const int* in_sizes, int n_in,
                              void* d_out, int out_size, void* d_ws, size_t ws_size,
                              hipStream_t stream) {
  (void)in_sizes; (void)n_in; (void)out_size; (void)ws_size;
  const float* pred = (const float*)d_in[0];
  const float* targ = (const float*)d_in[1];
  float* out = (float*)d_out;

  char* ws = (char*)d_ws;
  float* fA = (float*)ws;                                        // 64 MB
  float* fB = (float*)(ws + (size_t)NB * NPIX * sizeof(float));  // 64 MB
  char* aux = ws + 2ull * NB * NPIX * sizeof(float);
  double* sumd  = (double*)aux;          // [16]
  float*  rmin  = (float*)(sumd + NB);   // [16]
  float*  rmax  = rmin + NB;             // [16]
  float*  params = rmax + NB;            // [32] {mn, inv} per sample
  double* accum = (double*)(params + 2 * NB);

  const dim3 grid(NB * 512), block(256);

  init_state<<<1, 64, 0, stream>>>(rmin, rmax, sumd, params, accum);

  // k = 0: conv over bound=(pred-target)^2 computed on load
  erode_step<true, false><<<grid, block, 0, stream>>>(pred, targ, params, fA,
                                                      rmin, rmax, sumd);
  finalize_iter<<<1, 32, 0, stream>>>(rmin, rmax, sumd, params, accum, out, 0, 0);

  float* src = fA;
  float* dst = fB;
  for (int k = 1; k < 9; ++k) {
    erode_step<false, false><<<grid, block, 0, stream>>>(src, nullptr, params, dst,
                                                         rmin, rmax, sumd);
    finalize_iter<<<1, 32, 0, stream>>>(rmin, rmax, sumd, params, accum, out, k, 0);
    float* t = src; src = dst; dst = t;
  }
  // k = 9: field output never consumed -> skip the 64 MB store
  erode_step<false, true><<<grid, block, 0, stream>>>(src, nullptr, params, dst,
                                                      rmin, rmax, sumd);
  finalize_iter<<<1, 32, 0, stream>>>(rmin, rmax, sumd, params, accum, out, 9, 1);
}